// RotaryPositionalEmbedding_30245159699069
// MI455X (gfx1250) — compile-verified
//
#include <hip/hip_runtime.h>
#include <hip/hip_bf16.h>

// RoPE, x:[4,4096,2048] f32. Memory-bound streaming kernel:
//  - B128 vectorized load/store (native ext_vector_type so NT builtins accept them)
//  - NT stores for output (keep 128MB input resident in the 192MB L2 across replays)
//  - global_prefetch one stride ahead
//  - hardware v_sin/v_cos for the angles (cost hidden under HBM latency)

#define ROPE_DIM        2048
#define ROPE_SEQ        4096
#define F4_PER_ROW      512   // 2048 / 4
#define ROT_F4_PER_ROW  256   // 1024 / 4  (wave-aligned split: 256 % 32 == 0)

typedef float v4f __attribute__((ext_vector_type(4)));
typedef float v2f __attribute__((ext_vector_type(2)));

__global__ __launch_bounds__(256)
void RotaryPositionalEmbedding_30245159699069_kernel(const float* __restrict__ x,
                                                     const float* __restrict__ freqs,
                                                     float* __restrict__ out,
                                                     int total_f4) {
    const v4f* __restrict__ x4 = reinterpret_cast<const v4f*>(x);
    v4f* __restrict__ o4       = reinterpret_cast<v4f*>(out);
    const v2f* __restrict__ f2 = reinterpret_cast<const v2f*>(freqs);

    const int stride = gridDim.x * blockDim.x;

    for (int g = blockIdx.x * blockDim.x + threadIdx.x; g < total_f4; g += stride) {
        // Prefetch next grid-stride iteration's cacheline (global_prefetch_b8).
        if (g + stride < total_f4)
            __builtin_prefetch(&x4[g + stride], 0, 0);

        const int row = g >> 9;          // g / F4_PER_ROW
        const int c4  = g & (F4_PER_ROW - 1);

        const v4f v = x4[g];             // global_load_b128

        if (c4 < ROT_F4_PER_ROW) {
            // Rotary half: this float4 holds pairs (2*c4, 2*c4+1).
            const int  pos = row & (ROPE_SEQ - 1);
            const v2f  f   = f2[c4];     // freqs[2*c4], freqs[2*c4+1] (global_load_b64)
            const float p  = (float)pos;

            float s0, c0, s1, c1;
            __sincosf(p * f.x, &s0, &c0);   // v_sin_f32 / v_cos_f32
            __sincosf(p * f.y, &s1, &c1);

            v4f r;
            r.x = v.x * c0 - v.y * s0;
            r.y = v.x * s0 + v.y * c0;
            r.z = v.z * c1 - v.w * s1;
            r.w = v.z * s1 + v.w * c1;

            __builtin_nontemporal_store(r, &o4[g]);   // TH=NT global_store_b128
        } else {
            // Pass-through half: pure streaming copy.
            __builtin_nontemporal_store(v, &o4[g]);
        }
    }
}

extern "C" void kernel_launch(void* const* d_in, const int* in_sizes, int n_in,
                              void* d_out, int out_size, void* d_ws, size_t ws_size,
                              hipStream_t stream) {
    (void)n_in; (void)d_ws; (void)ws_size; (void)out_size;

    const float* x     = (const float*)d_in[0];   // [4, 4096, 2048] f32
    const float* freqs = (const float*)d_in[1];   // [512] f32
    float* out         = (float*)d_out;

    const int total_f4 = in_sizes[0] / 4;         // 8,388,608 float4s

    const int threads = 256;
    // ~4 grid-stride iterations per thread: plenty of blocks to fill all WGPs,
    // enough work per thread for the prefetch to be useful.
    int blocks = (total_f4 + threads * 4 - 1) / (threads * 4);   // 8192

    RotaryPositionalEmbedding_30245159699069_kernel<<<blocks, threads, 0, stream>>>(
        x, freqs, out, total_f4);
}